// InterScaleBlock_26946624815680
// MI455X (gfx1250) — compile-verified
//
#include <hip/hip_runtime.h>
#include <hip/hip_bf16.h>

#define H 256
#define N0 200000
#define N1 20000
#define N2 2000
#define LN_EPS 1e-5f
#define LDSP (H + 4)   // +4 floats: rows stay 16B-aligned AND column reads are bank-conflict-free

typedef float v2f __attribute__((ext_vector_type(2)));
typedef float v4f __attribute__((ext_vector_type(4)));
typedef float v8f __attribute__((ext_vector_type(8)));

#if __has_builtin(__builtin_amdgcn_global_load_async_to_lds_b128) && \
    __has_builtin(__builtin_amdgcn_s_wait_asynccnt)
#define USE_ASYNC_LDS 1
typedef int ab128 __attribute__((vector_size(16)));             // V4i payload type
typedef __attribute__((address_space(1))) ab128* g_b128p;       // global int4*
typedef __attribute__((address_space(3))) ab128* l_b128p;       // LDS int4*
#endif

// ---------------------------------------------------------------------------
// Scatter: fp32 hardware atomics into L2-resident pooled accumulator.
// Regular-temporal loads: h0 lines left in L2 are reused by gather_ln later.
// ---------------------------------------------------------------------------
__global__ __launch_bounds__(256) void scatter_add_kernel(
    const float* __restrict__ src, const int* __restrict__ idx,
    float* __restrict__ pool, int n)
{
    long tid = (long)blockIdx.x * blockDim.x + threadIdx.x;  // over n*64 (float4 each)
    if (tid >= (long)n * (H / 4)) return;
    int i = (int)(tid >> 6);
    int c = ((int)tid & 63) * 4;
    v4f v = *(const v4f*)(src + (long)i * H + c);
    float* p = pool + (long)idx[i] * H + c;
    atomicAdd(p + 0, v.x);
    atomicAdd(p + 1, v.y);
    atomicAdd(p + 2, v.z);
    atomicAdd(p + 3, v.w);
}

__global__ __launch_bounds__(256) void count_kernel(
    const int* __restrict__ idx, float* __restrict__ cnt, int n)
{
    int i = blockIdx.x * blockDim.x + threadIdx.x;
    if (i < n) atomicAdd(cnt + idx[i], 1.0f);
}

// ---------------------------------------------------------------------------
// Fused tile kernel: msg = (X .* rowScale) @ W^T ;  then either
//   mode 0:  out = msg + bias
//   mode 1:  out = LayerNorm(hin + msg + bias) * gamma + beta
// Block = 16 rows x 256 cols, 128 threads (4 waves), WMMA f32 16x16x4.
// nrows must be a multiple of 16 (20000 and 2000 both are).
// ---------------------------------------------------------------------------
__global__ __launch_bounds__(128) void gemm_tile_kernel(
    const float* __restrict__ X, const float* __restrict__ cnt,
    const float* __restrict__ W, const float* __restrict__ bias,
    const float* __restrict__ hin, const float* __restrict__ gamma,
    const float* __restrict__ beta, float* __restrict__ out, int mode)
{
    __shared__ float lds[16][LDSP];
    __shared__ float red_s[16][8];
    __shared__ float red_q[16][8];
    __shared__ float row_mu[16];
    __shared__ float row_rs[16];

    const int t  = threadIdx.x;
    const int r  = t >> 3;             // 0..15  (row within tile)
    const int c0 = (t & 7) * 32;       // 32-col strip per thread
    const long row0 = (long)blockIdx.x * 16;

    // Phase 0: stage X tile into LDS (optionally scaled by 1/max(count,1))
    float scale = 1.0f;
    if (cnt) scale = 1.0f / fmaxf(cnt[row0 + r], 1.0f);

#ifdef USE_ASYNC_LDS
    {
        const float* src = X + (row0 + r) * H + c0;
        #pragma unroll
        for (int j = 0; j < 32; j += 4) {
            // AS(1) value == generic 64-bit value; AS(3) offset == low 32 bits
            g_b128p ga = (g_b128p)(unsigned long long)(src + j);
            l_b128p la = (l_b128p)(unsigned int)(unsigned long long)(&lds[r][c0 + j]);
            __builtin_amdgcn_global_load_async_to_lds_b128(ga, la, 0, 0);
        }
        __builtin_amdgcn_s_wait_asynccnt(0);
    }
    __syncthreads();
    if (cnt) {   // rescale in place (each thread owns the region it loaded)
        #pragma unroll
        for (int j = 0; j < 32; j++) lds[r][c0 + j] *= scale;
        __syncthreads();
    }
#else
    {
        const float* src = X + (row0 + r) * H + c0;
        #pragma unroll
        for (int j = 0; j < 32; j += 4) {
            v4f v = *(const v4f*)(src + j);
            lds[r][c0 + j + 0] = v.x * scale;
            lds[r][c0 + j + 1] = v.y * scale;
            lds[r][c0 + j + 2] = v.z * scale;
            lds[r][c0 + j + 3] = v.w * scale;
        }
    }
    __syncthreads();
#endif

    // Phase 1: WMMA K-loop. Wave w owns 4 consecutive 16-col N-tiles.
    const int lane  = t & 31;
    const int wave  = t >> 5;           // 0..3
    const int m     = lane & 15;        // A row / B,D col within tile
    const int koff  = (lane >> 4) * 2;  // K sub-pair selected by lane half
    const int nbase = wave * 64;

    v8f acc[4] = {};                    // 4 accumulators, 16x16 f32 each

    const float* wrow[4];
    #pragma unroll
    for (int i = 0; i < 4; i++)
        wrow[i] = W + (size_t)(nbase + i * 16 + m) * H + koff;  // B[k][n] = W[n][k]

    for (int k = 0; k < H; k += 4) {
        v2f a;                                       // A: 16x4 tile, M=m
        a.x = lds[m][k + koff];
        a.y = lds[m][k + koff + 1];
        #pragma unroll
        for (int i = 0; i < 4; i++) {
            float2 wv = *(const float2*)(wrow[i] + k);
            v2f b; b.x = wv.x; b.y = wv.y;           // B: 4x16 tile of W^T
            acc[i] = __builtin_amdgcn_wmma_f32_16x16x4_f32(
                false, a, false, b, (short)0, acc[i], false, false);
        }
    }

    __syncthreads();   // X tile fully consumed; reuse LDS for the output tile

    // D layout: VGPR v -> row v + 8*(lane>>4), col = lane&15 (within tile)
    const int orow = (lane >> 4) * 8;
    #pragma unroll
    for (int i = 0; i < 4; i++) {
        const int ocol = nbase + i * 16 + m;
        #pragma unroll
        for (int v = 0; v < 8; v++)
            lds[orow + v][ocol] = acc[i][v];
    }
    __syncthreads();

    // Phase 2
    if (mode == 0) {
        float* dst = out + (row0 + r) * H + c0;
        #pragma unroll
        for (int j = 0; j < 32; j++)
            dst[j] = lds[r][c0 + j] + bias[c0 + j];
        return;
    }

    // mode 1: residual + LayerNorm (8 threads cooperate per row)
    float s = 0.0f, q = 0.0f;
    const float* hr = hin + (row0 + r) * H + c0;
    #pragma unroll
    for (int j = 0; j < 32; j++) {
        float x = lds[r][c0 + j] + bias[c0 + j] + hr[j];
        lds[r][c0 + j] = x;
        s += x; q += x * x;
    }
    red_s[r][t & 7] = s;
    red_q[r][t & 7] = q;
    __syncthreads();
    if ((t & 7) == 0) {
        float ss = 0.0f, qq = 0.0f;
        #pragma unroll
        for (int j = 0; j < 8; j++) { ss += red_s[r][j]; qq += red_q[r][j]; }
        float mu  = ss * (1.0f / H);
        float var = qq * (1.0f / H) - mu * mu;
        row_mu[r] = mu;
        row_rs[r] = rsqrtf(var + LN_EPS);
    }
    __syncthreads();
    const float mu = row_mu[r], rs = row_rs[r];
    float* dst = out + (row0 + r) * H + c0;
    #pragma unroll
    for (int j = 0; j < 32; j++)
        dst[j] = (lds[r][c0 + j] - mu) * rs * gamma[c0 + j] + beta[c0 + j];
}

// ---------------------------------------------------------------------------
// Bandwidth kernel: out[i] = LayerNorm(base[i] + Y[idx[i]]); 1 wave per row.
// base is last-use -> non-temporal loads; out is write-once -> non-temporal
// stores (protects L2 residency of the gathered Y table and cached base rows).
// ---------------------------------------------------------------------------
__global__ __launch_bounds__(256) void gather_ln_kernel(
    const float* __restrict__ base, const float* __restrict__ Y,
    const int* __restrict__ idx, const float* __restrict__ gamma,
    const float* __restrict__ beta, float* __restrict__ out, int n)
{
    const int lane = threadIdx.x & 31;
    const int w    = threadIdx.x >> 5;
    const long row = (long)blockIdx.x * 8 + w;
    if (row >= n) return;

    const long srow = (long)idx[row] * H;
    const float* br = base + row * H + lane * 8;
    const float* yr = Y + srow + lane * 8;

    v4f b0 = __builtin_nontemporal_load((const v4f*)(br));
    v4f b1 = __builtin_nontemporal_load((const v4f*)(br + 4));
    v4f y0 = *(const v4f*)(yr);          // gathered table: keep cacheable
    v4f y1 = *(const v4f*)(yr + 4);

    float x[8];
    x[0] = b0.x + y0.x; x[1] = b0.y + y0.y; x[2] = b0.z + y0.z; x[3] = b0.w + y0.w;
    x[4] = b1.x + y1.x; x[5] = b1.y + y1.y; x[6] = b1.z + y1.z; x[7] = b1.w + y1.w;

    float s = 0.0f, q = 0.0f;
    #pragma unroll
    for (int j = 0; j < 8; j++) { s += x[j]; q += x[j] * x[j]; }

    // wave32 butterfly reduction
    #pragma unroll
    for (int off = 16; off > 0; off >>= 1) {
        s += __shfl_xor(s, off, 32);
        q += __shfl_xor(q, off, 32);
    }
    const float mu  = s * (1.0f / H);
    const float var = q * (1.0f / H) - mu * mu;
    const float rs  = rsqrtf(var + LN_EPS);

    v4f g0 = *(const v4f*)(gamma + lane * 8);
    v4f g1 = *(const v4f*)(gamma + lane * 8 + 4);
    v4f e0 = *(const v4f*)(beta + lane * 8);
    v4f e1 = *(const v4f*)(beta + lane * 8 + 4);

    v4f o0, o1;
    o0.x = (x[0] - mu) * rs * g0.x + e0.x;
    o0.y = (x[1] - mu) * rs * g0.y + e0.y;
    o0.z = (x[2] - mu) * rs * g0.z + e0.z;
    o0.w = (x[3] - mu) * rs * g0.w + e0.w;
    o1.x = (x[4] - mu) * rs * g1.x + e1.x;
    o1.y = (x[5] - mu) * rs * g1.y + e1.y;
    o1.z = (x[6] - mu) * rs * g1.z + e1.z;
    o1.w = (x[7] - mu) * rs * g1.w + e1.w;

    float* dst = out + row * H + lane * 8;
    __builtin_nontemporal_store(o0, (v4f*)(dst));
    __builtin_nontemporal_store(o1, (v4f*)(dst + 4));
}

// ---------------------------------------------------------------------------
extern "C" void kernel_launch(void* const* d_in, const int* in_sizes, int n_in,
                              void* d_out, int out_size, void* d_ws, size_t ws_size,
                              hipStream_t stream) {
    const float* h0    = (const float*)d_in[0];
    const float* h1    = (const float*)d_in[1];
    const float* h2    = (const float*)d_in[2];
    const int*   a0    = (const int*)d_in[3];
    const int*   a1    = (const int*)d_in[4];
    const float* Wup01 = (const float*)d_in[5];
    const float* bup01 = (const float*)d_in[6];
    const float* Wdn10 = (const float*)d_in[7];
    const float* bdn10 = (const float*)d_in[8];
    const float* Wup12 = (const float*)d_in[9];
    const float* bup12 = (const float*)d_in[10];
    const float* Wdn21 = (const float*)d_in[11];
    const float* bdn21 = (const float*)d_in[12];
    const float* g0    = (const float*)d_in[13];
    const float* be0   = (const float*)d_in[14];
    const float* g1    = (const float*)d_in[15];
    const float* be1   = (const float*)d_in[16];
    const float* g2    = (const float*)d_in[17];
    const float* be2   = (const float*)d_in[18];

    float* out0 = (float*)d_out;                  // h0 result [N0,H]
    float* out1 = out0 + (size_t)N0 * H;          // h1 result [N1,H]
    float* out2 = out1 + (size_t)N1 * H;          // h2 result [N2,H]

    float* ws      = (float*)d_ws;
    float* pooled0 = ws;                          // [N1,H]
    float* cnt0    = pooled0 + (size_t)N1 * H;    // [N1]
    float* h1a     = cnt0 + N1;                   // [N1,H]  h1 after step 1
    float* Y1      = h1a + (size_t)N1 * H;        // [N1,H]  h1a @ Wdn10^T + b
    float* pooled1 = Y1 + (size_t)N1 * H;         // [N2,H]
    float* cnt1    = pooled1 + (size_t)N2 * H;    // [N2]
    float* Y2      = cnt1 + N2;                   // [N2,H]  h2' @ Wdn21^T + b

    // ---- Step 1: pooled0 = scatter_mean(h0, assign0); h1a = LN(h1 + lin) ----
    (void)hipMemsetAsync(pooled0, 0, ((size_t)N1 * H + N1) * sizeof(float), stream);
    (void)hipMemsetAsync(pooled1, 0, ((size_t)N2 * H + N2) * sizeof(float), stream);

    {
        long elems = (long)N0 * (H / 4);
        scatter_add_kernel<<<(int)((elems + 255) / 256), 256, 0, stream>>>(h0, a0, pooled0, N0);
        count_kernel<<<(N0 + 255) / 256, 256, 0, stream>>>(a0, cnt0, N0);
    }
    gemm_tile_kernel<<<N1 / 16, 128, 0, stream>>>(
        pooled0, cnt0, Wup01, bup01, h1, g1, be1, h1a, 1);

    // ---- Step 2: h0' = LN(h0 + (h1a @ Wdn10^T + b)[assign0]) ----
    gemm_tile_kernel<<<N1 / 16, 128, 0, stream>>>(
        h1a, nullptr, Wdn10, bdn10, nullptr, nullptr, nullptr, Y1, 0);
    gather_ln_kernel<<<N0 / 8, 256, 0, stream>>>(h0, Y1, a0, g0, be0, out0, N0);

    // ---- Step 3: pooled1 = scatter_mean(h1a, assign1); h2' = LN(h2 + lin) ----
    {
        long elems = (long)N1 * (H / 4);
        scatter_add_kernel<<<(int)((elems + 255) / 256), 256, 0, stream>>>(h1a, a1, pooled1, N1);
        count_kernel<<<(N1 + 255) / 256, 256, 0, stream>>>(a1, cnt1, N1);
    }
    gemm_tile_kernel<<<N2 / 16, 128, 0, stream>>>(
        pooled1, cnt1, Wup12, bup12, h2, g2, be2, out2, 1);

    // ---- Step 4: h1' = LN(h1a + (h2' @ Wdn21^T + b)[assign1]) ----
    gemm_tile_kernel<<<N2 / 16, 128, 0, stream>>>(
        out2, nullptr, Wdn21, bdn21, nullptr, nullptr, nullptr, Y2, 0);
    gather_ln_kernel<<<N1 / 8, 256, 0, stream>>>(h1a, Y2, a1, g1, be1, out1, N1);
}